// VectorQuantizer_28784870817819
// MI455X (gfx1250) — compile-verified
//
#include <hip/hip_runtime.h>

// VQ: x (64,64,32,32) f32, codebook (512,64) f32.
// d_out = [ loss(1) | quantized(64*64*32*32) | idx(64*1024) as float ]
//
// Fully fused: TDM stages the codebook into LDS once per block, each wave
// computes a 16-row tile of x @ codebook^T via V_WMMA_F32_16X16X4_F32,
// does the argmin in-register, then (reusing the A fragments that already
// hold this wave's x tile) writes the quantized output and the loss partial
// without re-reading x.  Traffic ~= 16MB x read + 16MB q write  (~1.4us
// at 23.3 TB/s HBM; the GEMM itself is only 4.3 GFLOP and is hidden).

typedef float v2f __attribute__((ext_vector_type(2)));
typedef float v8f __attribute__((ext_vector_type(8)));
typedef unsigned int u32x4 __attribute__((ext_vector_type(4)));
typedef int i32x4 __attribute__((ext_vector_type(4)));
typedef int i32x8 __attribute__((ext_vector_type(8)));

#define K_ENTRIES 512
#define DIM       64
#define HW        1024          // 32*32
#define N_ROWS    65536         // 64 * HW
#define QELEMS    4194304       // 64*64*32*32
#define NBLOCKS   512           // N_ROWS / (8 waves * 16 rows)

// ---------------------------------------------------------------------------
// Fused kernel: TDM codebook stage -> WMMA distances -> argmin -> quantize+loss
// ---------------------------------------------------------------------------
__global__ __launch_bounds__(256) void vq_fused_kernel(
    const float* __restrict__ x, const float* __restrict__ cb,
    float* __restrict__ out_q, float* __restrict__ out_idx,
    float* __restrict__ partial)
{
    __shared__ float cb_lds[K_ENTRIES * DIM];   // 128 KB staged codebook
    __shared__ float e2h[K_ENTRIES];            // 0.5 * ||e_k||^2
    __shared__ int   idxl[8 * 16];              // per-wave winning indices
    __shared__ float wred[8];                   // per-wave loss partials

    const int t = threadIdx.x;

    // ---- Stage codebook into LDS with the Tensor Data Mover (wave 0 only;
    //      TDM ignores EXEC, issues once per wave). 2-D D# per ISA ch.8. ----
    if ((t >> 5) == 0) {
        const unsigned long long ga = (unsigned long long)(uintptr_t)cb;
        const unsigned int lds_off = (unsigned int)(uintptr_t)(&cb_lds[0]);
        u32x4 g0;
        g0[0] = 1u;                                      // count=1, user mode
        g0[1] = lds_off;                                 // lds_addr [63:32]
        g0[2] = (unsigned int)(ga & 0xffffffffull);      // global_addr lo
        g0[3] = (unsigned int)((ga >> 32) & 0x1ffffffull)// global_addr[56:32]
                | 0x80000000u;                           // type=2 ("image")
        i32x8 g1;
        g1[0] = (int)(2u << 16);          // data_size=4B, no mask/pad/iterate
        g1[1] = (int)(64u << 16);         // tensor_dim0 = 64   (bits 63:48)
        g1[2] = (int)(512u << 16);        // tensor_dim1 = 512  (bits 95:80)
        g1[3] = (int)(64u << 16);         // tile_dim0 = 64     (bits 127:112)
        g1[4] = (int)512;                 // tile_dim1 = 512    (bits 143:128)
        g1[5] = (int)64;                  // tensor_dim0_stride = 64
        g1[6] = 0;
        g1[7] = 0;
        i32x4 gz4 = {0, 0, 0, 0};         // groups 2/3 unused (2-D tensor)
        i32x8 gz8 = {0, 0, 0, 0, 0, 0, 0, 0};
        __builtin_amdgcn_tensor_load_to_lds(g0, g1, gz4, gz4, gz8, 0);
        __builtin_amdgcn_s_wait_tensorcnt(0);
    }
    __syncthreads();

    // ---- Precompute 0.5*||e_k||^2 from the staged codebook ----
    for (int e = t; e < K_ENTRIES; e += 256) {
        const float* row = &cb_lds[e * DIM];
        float s = 0.f;
        #pragma unroll
        for (int k = 0; k < DIM; ++k) s += row[k] * row[k];
        e2h[e] = 0.5f * s;
    }
    __syncthreads();

    const int wave = t >> 5;
    const int lane = t & 31;
    const int half = lane >> 4;     // 0: lanes 0-15, 1: lanes 16-31
    const int mrow = lane & 15;
    const int rowBase = (blockIdx.x * 8 + wave) * 16;
    const int row = rowBase + mrow;
    const int b      = rowBase >> 10;        // whole tile shares batch index
    const int hw     = row & (HW - 1);
    const float* xrow = x + (size_t)b * (DIM * HW) + hw; // element d at +d*HW

    // A fragments: 16x4 f32 tile per k-step (lane holds K=k0,k0+1 of its row;
    // half 0 -> k0 = 4k, half 1 -> k0 = 4k+2 per the ISA 16x4 f32 A layout).
    v2f A[16];
    #pragma unroll
    for (int ks = 0; ks < 16; ++ks) {
        const int k0 = ks * 4 + half * 2;
        A[ks][0] = xrow[(size_t)(k0    ) * HW];
        A[ks][1] = xrow[(size_t)(k0 + 1) * HW];
    }

    // ---- distance GEMM + running argmin ----
    float bestS[8];
    int   bestI[8];
    #pragma unroll
    for (int v = 0; v < 8; ++v) { bestS[v] = 3.0e38f; bestI[v] = 0; }

    for (int tile = 0; tile < K_ENTRIES / 16; ++tile) {
        const int entry = tile * 16 + mrow;     // candidate at this N position
        const float* crow = &cb_lds[entry * DIM];
        v8f c = {0.f, 0.f, 0.f, 0.f, 0.f, 0.f, 0.f, 0.f};
        #pragma unroll
        for (int ks = 0; ks < 16; ++ks) {
            const int k0 = ks * 4 + half * 2;
            v2f bf;
            bf[0] = crow[k0];
            bf[1] = crow[k0 + 1];
            c = __builtin_amdgcn_wmma_f32_16x16x4_f32(
                    false, A[ks], false, bf, (short)0, c, false, false);
        }
        const float e2 = e2h[entry];
        #pragma unroll
        for (int v = 0; v < 8; ++v) {
            const float s = e2 - c[v];          // argmin-equivalent score
            if (s < bestS[v]) { bestS[v] = s; bestI[v] = entry; }
        }
    }

    // Min-reduce across the 16 lanes of each half (masks < 16 stay in-half).
    #pragma unroll
    for (int m = 1; m < 16; m <<= 1) {
        #pragma unroll
        for (int v = 0; v < 8; ++v) {
            const float os = __shfl_xor(bestS[v], m, 32);
            const int   oi = __shfl_xor(bestI[v], m, 32);
            if (os < bestS[v] || (os == bestS[v] && oi < bestI[v])) {
                bestS[v] = os; bestI[v] = oi;
            }
        }
    }

    // Publish winners: D-layout slot v = row v (half 0) / row v+8 (half 1).
    if (mrow == 0) {
        #pragma unroll
        for (int v = 0; v < 8; ++v) {
            idxl[wave * 16 + half * 8 + v] = bestI[v];
            out_idx[rowBase + half * 8 + v] = (float)bestI[v];
        }
    }
    __syncthreads();

    // ---- Quantize + loss epilogue: reuse A fragments (x already in VGPRs).
    //      Lane (r) and lane (r+16) split row r's 64 dims (same k0 pattern).
    const int id = idxl[wave * 16 + mrow];
    const float* qrow = &cb_lds[id * DIM];
    float* oq = out_q + (size_t)b * (DIM * HW) + hw;
    float ls = 0.f;
    #pragma unroll
    for (int ks = 0; ks < 16; ++ks) {
        const int k0 = ks * 4 + half * 2;
        const float q0 = qrow[k0];
        const float q1 = qrow[k0 + 1];
        oq[(size_t)(k0    ) * HW] = q0;
        oq[(size_t)(k0 + 1) * HW] = q1;
        const float d0 = q0 - A[ks][0];
        const float d1 = q1 - A[ks][1];
        ls += d0 * d0 + d1 * d1;
    }
    // wave-sum (all 32 lanes together cover the 16x64 tile exactly once)
    #pragma unroll
    for (int m = 1; m < 32; m <<= 1) ls += __shfl_xor(ls, m, 32);
    if (lane == 0) wred[wave] = ls;
    __syncthreads();
    if (t == 0) {
        float s = 0.f;
        #pragma unroll
        for (int w = 0; w < 8; ++w) s += wred[w];
        partial[blockIdx.x] = s;
    }
}

// ---------------------------------------------------------------------------
// Final loss = 1.25 * mean((q-x)^2)   (q_latent == e_latent numerically)
// ---------------------------------------------------------------------------
__global__ __launch_bounds__(256) void vq_loss_kernel(
    const float* __restrict__ partial, float* __restrict__ out_loss)
{
    __shared__ float red[256];
    const int t = threadIdx.x;
    float s = 0.f;
    for (int i = t; i < NBLOCKS; i += 256) s += partial[i];
    red[t] = s;
    __syncthreads();
    #pragma unroll
    for (int k = 128; k > 0; k >>= 1) {
        if (t < k) red[t] += red[t + k];
        __syncthreads();
    }
    if (t == 0) out_loss[0] = red[0] * (1.25f / (float)QELEMS);
}

extern "C" void kernel_launch(void* const* d_in, const int* in_sizes, int n_in,
                              void* d_out, int out_size, void* d_ws, size_t ws_size,
                              hipStream_t stream) {
    const float* x  = (const float*)d_in[0];   // (64,64,32,32)
    const float* cb = (const float*)d_in[1];   // (512,64)

    float* out      = (float*)d_out;
    float* out_loss = out;
    float* out_q    = out + 1;
    float* out_idx  = out + 1 + (size_t)QELEMS;

    float* partial = (float*)d_ws;             // 512 floats

    vq_fused_kernel<<<NBLOCKS, 256, 0, stream>>>(x, cb, out_q, out_idx, partial);
    vq_loss_kernel<<<1, 256, 0, stream>>>(partial, out_loss);
}